// IoU3DLoss_77876347011662
// MI455X (gfx1250) — compile-verified
//
#include <hip/hip_runtime.h>
#include <hip/hip_bf16.h>

// ---------------------------------------------------------------------------
// 3D rotated-box IoU loss for MI455X (gfx1250, wave32).
//
// Compute-bound f32 VALU kernel (bytes ~15.7MB -> 0.7us @ 23.3TB/s; ~7k f32
// ops/pair dominate). CDNA5-oriented structure:
//  * One thread per box pair. The 24 candidate polygon vertices live in LDS
//    with a [slot][tid] layout (4B * tid stride -> bank-conflict-free on the
//    64-bank LDS); validity is a 24-bit register mask. 36.9KB/block of the
//    320KB WGP LDS pool -> ~32 resident waves per WGP.
//  * Sort key = diamond pseudo-angle y/(|x|+|y|) + quadrant fixup: strictly
//    monotone in atan2 (period 4), so the cyclic vertex order and shoelace
//    area match the reference argsort -- no OCML atan2 (30+ instr) needed.
//  * Polygon area via a BRANCHLESS O(24^2) angular-successor scan: invalid
//    slots hold key=1e30 so the min-positive-delta search never picks them;
//    contributions are gated with one cndmask. No per-lane divergent
//    branches in the hot loop (each lane has a different mask, so skipping
//    would only churn EXEC). Inner 24-iter loop unrolled (constant ds_load
//    offsets), outer loop rolled to keep the kernel I$-resident.
//    (A fully-unrolled register sorting network was tried and rejected: the
//    compiler emitted ~44k straight-line VALU ops for it.)
//  * Deterministic reduction: per-block LDS tree -> d_ws partials -> one
//    wave32 accumulates in fixed lane-strided order and collapses its 32
//    lane partials with one exact f32 WMMA (V_WMMA_F32_16X16X4_F32,
//    A = lane partials, B = ones => D rows are acc_m + acc_{m+16}).
// ---------------------------------------------------------------------------

#define EPS_F 1e-8f
#define BLK 128
#define NSLOT 24
#define PERIOD_F 4.0f

typedef float v2f __attribute__((ext_vector_type(2)));
typedef float v8f __attribute__((ext_vector_type(8)));

__device__ __forceinline__ float diamond_key(float x, float y) {
    // Monotone-in-angle pseudo-angle, period 4, range [-1, 3).
    float den = fabsf(x) + fabsf(y);
    float t = y / fmaxf(den, 1e-30f);
    return (x >= 0.0f) ? t : 2.0f - t;
}

__global__ void __launch_bounds__(BLK)
iou3d_loss_kernel(const float* __restrict__ pred,
                  const float* __restrict__ target,
                  const float* __restrict__ weight,
                  float* __restrict__ partials,
                  int N) {
    __shared__ float sX[NSLOT * BLK];
    __shared__ float sY[NSLOT * BLK];
    __shared__ float sK[NSLOT * BLK];
    __shared__ float sRed[BLK];

    const int tid = threadIdx.x;
    const int i   = blockIdx.x * BLK + tid;
#define SLOT(s) ((s) * BLK + tid)

    float loss = 0.0f;
    if (i < N) {
        const float* p = pred   + (size_t)i * 7;
        const float* q = target + (size_t)i * 7;
        __builtin_prefetch(q, 0, 0);   // global_prefetch_b8

        float p0 = p[0], p1 = p[1], p2 = p[2], p3 = p[3], p4 = p[4], p5 = p[5], p6 = p[6];
        float q0 = q[0], q1 = q[1], q2 = q[2], q3 = q[3], q4 = q[4], q5 = q[5], q6 = q[6];

        float cA, sA, cB, sB;
        sincosf(p6, &sA, &cA);
        sincosf(q6, &sB, &cB);

        // Box corners, local order: (-.5,-.5),(.5,-.5),(.5,.5),(-.5,.5)
        const float lox[4] = {-0.5f, 0.5f, 0.5f, -0.5f};
        const float loy[4] = {-0.5f, -0.5f, 0.5f, 0.5f};
        float c1x[4], c1y[4], c2x[4], c2y[4];
#pragma unroll
        for (int k = 0; k < 4; ++k) {
            float sx = lox[k] * p3, sy = loy[k] * p4;
            c1x[k] = sx * cA - sy * sA + p0;
            c1y[k] = sx * sA + sy * cA + p1;
            float tx = lox[k] * q3, ty = loy[k] * q4;
            c2x[k] = tx * cB - ty * sB + q0;
            c2y[k] = tx * sB + ty * cB + q1;
        }

        unsigned mask = 0u;
        float sumx = 0.f, sumy = 0.f;
        int nv = 0;

        // 16 edge-edge intersections -> slots 0..15
#pragma unroll
        for (int ii = 0; ii < 4; ++ii) {
            float p1x = c1x[ii], p1y = c1y[ii];
            float d1x = c1x[(ii + 1) & 3] - p1x;
            float d1y = c1y[(ii + 1) & 3] - p1y;
#pragma unroll
            for (int jj = 0; jj < 4; ++jj) {
                float q1x = c2x[jj], q1y = c2y[jj];
                float d2x = c2x[(jj + 1) & 3] - q1x;
                float d2y = c2y[(jj + 1) & 3] - q1y;
                float den = d1x * d2y - d1y * d2x;
                bool ok = fabsf(den) > EPS_F;
                float rden = 1.0f / (ok ? den : 1.0f);
                float rx = q1x - p1x, ry = q1y - p1y;
                float t  = (rx * d2y - ry * d2x) * rden;
                float ss = (rx * d1y - ry * d1x) * rden;
                bool valid = ok && (t >= 0.f) && (t <= 1.f) && (ss >= 0.f) && (ss <= 1.f);
                float ix = p1x + t * d1x;
                float iy = p1y + t * d1y;
                int slot = ii * 4 + jj;
                sX[SLOT(slot)] = ix;
                sY[SLOT(slot)] = iy;
                if (valid) { mask |= 1u << slot; sumx += ix; sumy += iy; ++nv; }
            }
        }
        // Corners of box1 inside box2 -> slots 16..19
#pragma unroll
        for (int k = 0; k < 4; ++k) {
            float dx = c1x[k] - q0, dy = c1y[k] - q1;
            float lx = dx * cB + dy * sB;
            float ly = -dx * sB + dy * cB;
            bool valid = (fabsf(lx) <= q3 * 0.5f) && (fabsf(ly) <= q4 * 0.5f);
            int slot = 16 + k;
            sX[SLOT(slot)] = c1x[k];
            sY[SLOT(slot)] = c1y[k];
            if (valid) { mask |= 1u << slot; sumx += c1x[k]; sumy += c1y[k]; ++nv; }
        }
        // Corners of box2 inside box1 -> slots 20..23
#pragma unroll
        for (int k = 0; k < 4; ++k) {
            float dx = c2x[k] - p0, dy = c2y[k] - p1;
            float lx = dx * cA + dy * sA;
            float ly = -dx * sA + dy * cA;
            bool valid = (fabsf(lx) <= p3 * 0.5f) && (fabsf(ly) <= p4 * 0.5f);
            int slot = 20 + k;
            sX[SLOT(slot)] = c2x[k];
            sY[SLOT(slot)] = c2y[k];
            if (valid) { mask |= 1u << slot; sumx += c2x[k]; sumy += c2y[k]; ++nv; }
        }

        float cnt = (float)(nv > 1 ? nv : 1);
        float cx = sumx / cnt, cy = sumy / cnt;

        // Centroid-relative coords + monotone keys (branchless; invalid=1e30).
#pragma unroll
        for (int s = 0; s < NSLOT; ++s) {
            float x = sX[SLOT(s)] - cx;
            float y = sY[SLOT(s)] - cy;
            sX[SLOT(s)] = x;
            sY[SLOT(s)] = y;
            bool v = (mask >> s) & 1u;
            sK[SLOT(s)] = v ? diamond_key(x, y) : 1e30f;
        }

        // Branchless angular-successor shoelace: for each valid vertex, find
        // the valid vertex with the smallest positive key delta (mod period)
        // and accumulate the cross product. Cyclic successor chain == sorted
        // cyclic polygon, so this equals the reference's sorted shoelace.
        float area2 = 0.f;
#pragma unroll 1
        for (int s0 = 0; s0 < NSLOT; ++s0) {
            float ai = sK[SLOT(s0)];
            float xi = sX[SLOT(s0)];
            float yi = sY[SLOT(s0)];
            float best = 1e30f, bxv = xi, byv = yi;
#pragma unroll
            for (int s1 = 0; s1 < NSLOT; ++s1) {
                float aj = sK[SLOT(s1)];
                float xj = sX[SLOT(s1)];
                float yj = sY[SLOT(s1)];
                float d = aj - ai;
                d = (d <= 0.f) ? d + PERIOD_F : d;   // invalid aj stays huge
                bool take = d < best;
                best = take ? d : best;
                bxv = take ? xj : bxv;
                byv = take ? yj : byv;
            }
            float cr = xi * byv - yi * bxv;
            area2 += ((mask >> s0) & 1u) ? cr : 0.f;
        }
        float inter2d = (nv >= 3) ? 0.5f * fabsf(area2) : 0.f;

        // z-overlap and 3D IoU
        float zmax1 = p2 + 0.5f * p5, zmin1 = p2 - 0.5f * p5;
        float zmax2 = q2 + 0.5f * q5, zmin2 = q2 - 0.5f * q5;
        float zov = fmaxf(fminf(zmax1, zmax2) - fmaxf(zmin1, zmin2), 0.f);
        float inter3d = inter2d * zov;
        float v1 = p3 * p4 * p5;
        float v2 = q3 * q4 * q5;
        float uni = fmaxf(v1 + v2 - inter3d, EPS_F);
        loss = (1.0f - inter3d / uni) * weight[i];
    }

    // Deterministic per-block tree reduction.
    sRed[tid] = loss;
    __syncthreads();
#pragma unroll
    for (int s = BLK / 2; s > 0; s >>= 1) {
        if (tid < s) sRed[tid] += sRed[tid + s];
        __syncthreads();
    }
    if (tid == 0) partials[blockIdx.x] = sRed[0];
#undef SLOT
}

// Single-wave deterministic final reduction; 32 lane partials collapsed with
// one exact f32 WMMA (A = placed partials, B = ones => every D row holds
// acc_m + acc_{m+16}; lane0 + lane16 of the 8 D VGPRs give the grand total).
__global__ void iou3d_reduce_kernel(const float* __restrict__ partials,
                                    float* __restrict__ out,
                                    int nparts, float invN) {
    int lane = threadIdx.x;  // 32 threads, all active (EXEC all-ones for WMMA)
    float acc = 0.f;
    for (int k = lane; k < nparts; k += 32) acc += partials[k];

    v2f a; a[0] = acc; a[1] = 0.f;   // A[m][0]=acc_m, A[m][2]=acc_{m+16}
    v2f b; b[0] = 1.f; b[1] = 1.f;   // B = ones(4x16) under any lane layout
    v8f c = {};
    c = __builtin_amdgcn_wmma_f32_16x16x4_f32(false, a, false, b,
                                              (short)0, c, false, false);
    float t = c[0] + c[1] + c[2] + c[3] + c[4] + c[5] + c[6] + c[7];
    float total = __shfl(t, 0, 32) + __shfl(t, 16, 32);
    if (lane == 0) out[0] = total * invN;
}

extern "C" void kernel_launch(void* const* d_in, const int* in_sizes, int n_in,
                              void* d_out, int out_size, void* d_ws, size_t ws_size,
                              hipStream_t stream) {
    const float* pred   = (const float*)d_in[0];
    const float* target = (const float*)d_in[1];
    const float* weight = (const float*)d_in[2];
    float* out = (float*)d_out;
    float* ws  = (float*)d_ws;

    int N = in_sizes[0] / 7;              // 262144
    int nblocks = (N + BLK - 1) / BLK;    // 2048 partials

    iou3d_loss_kernel<<<nblocks, BLK, 0, stream>>>(pred, target, weight, ws, N);
    iou3d_reduce_kernel<<<1, 32, 0, stream>>>(ws, out, nblocks, 1.0f / (float)N);
}